// GraphConvolutionLayer_48584670053084
// MI455X (gfx1250) — compile-verified
//
#include <hip/hip_runtime.h>
#include <hip/hip_bf16.h>

// GAT layer for MI455X (gfx1250, wave32).
// B=4, N=512, F_IN=F_OUT=128.
//
// Kernel 1 (gat_proj): per 16-row tile of the flattened (B*N, F) matrix:
//   h_self = X@W_self+b, h_nb = X@W_nb+b  (WMMA f32 16x16x4, X tile in LDS)
//   s = h_self@A1_i, t = h_nb@A1_j        (chained through LDS, WMMA again)
// Kernel 2 (gat_attn): per (batch, 16-row) tile:
//   scores(i,j) = sum_k relu(s[i,k]+t[j,k]+a1b[k])*a2[k]  (VALU relu-dot, LDS tiles)
//   mask by adjacency, row softmax in LDS,
//   agg = attn @ h_nb via WMMA f32 16x16x4 (A-frags from LDS, K=512),
//   out = relu(h_self + agg).

typedef __attribute__((ext_vector_type(2))) float v2f;
typedef __attribute__((ext_vector_type(8))) float v8f;

#define NB 4
#define NN 512
#define FF 128

// Wave-level 16x16 tile GEMM: D = A(16xK, LDS, stride lstr) @ B(KxN global, row
// stride FF), output columns [n0, n0+16). WMMA f32 16x16x4 operand layout:
//   A vgpr0 = A[M=ln][k0+2*hi],   vgpr1 = A[M=ln][k0+2*hi+1]
//   B vgpr0 = B[k0+2*hi][n0+ln],  vgpr1 = B[k0+2*hi+1][n0+ln]
__device__ __forceinline__ v8f wave_gemm16(const float* __restrict__ lA, int lstr,
                                           const float* __restrict__ Bg,
                                           int n0, int ln, int hi, int K)
{
    v8f acc = {0.f, 0.f, 0.f, 0.f, 0.f, 0.f, 0.f, 0.f};
    for (int k0 = 0; k0 < K; k0 += 4) {
        int ka = k0 + 2 * hi;
        v2f a, b;
        a.x = lA[ln * lstr + ka];
        a.y = lA[ln * lstr + ka + 1];
        b.x = Bg[ka * FF + n0 + ln];
        b.y = Bg[(ka + 1) * FF + n0 + ln];
        acc = __builtin_amdgcn_wmma_f32_16x16x4_f32(false, a, false, b,
                                                    (short)0, acc, false, false);
    }
    return acc;
}

// Scatter a 16x16 D-fragment (+column bias) into LDS tile and global memory.
// D layout: vgpr v -> row (v + 8*hi), column (n0 + ln).
__device__ __forceinline__ void store_frag(v8f d, float* __restrict__ lT, int lstr,
                                           float* __restrict__ g, int grow0,
                                           const float* __restrict__ bias,
                                           int n0, int ln, int hi)
{
    float bv = bias ? bias[n0 + ln] : 0.f;
    #pragma unroll
    for (int v = 0; v < 8; ++v) {
        int r = v + 8 * hi;
        float val = d[v] + bv;
        if (lT) lT[r * lstr + n0 + ln] = val;
        g[(grow0 + r) * FF + n0 + ln] = val;
    }
}

__global__ __launch_bounds__(256) void gat_proj_kernel(
    const float* __restrict__ X,
    const float* __restrict__ Wself, const float* __restrict__ bself,
    const float* __restrict__ Wnb,   const float* __restrict__ bnb,
    const float* __restrict__ A1,    // (2F, F): rows 0..127 = A1_i, 128..255 = A1_j
    float* __restrict__ h_self, float* __restrict__ h_nb,
    float* __restrict__ s_out,  float* __restrict__ t_out)
{
    const int STR = 132;  // 128 + 4 pad: LDS banks (4*row + k) % 64, conflict-free
    __shared__ float lx[16 * STR];
    __shared__ float lhs[16 * STR];
    __shared__ float lhn[16 * STR];

    const int row0 = blockIdx.x * 16;   // 128 blocks cover B*N = 2048 rows
    const int tid  = threadIdx.x;

    for (int e = tid; e < 16 * FF; e += 256) {
        int r = e >> 7, c = e & 127;
        lx[r * STR + c] = X[(row0 + r) * FF + c];
    }
    __syncthreads();

    const int lane = tid & 31;
    const int ln = lane & 15, hi = lane >> 4;
    const int n0 = (tid >> 5) * 16;     // 8 waves x 16 cols = 128 output features

    // Stage 1: h_self / h_nb
    v8f hs = wave_gemm16(lx, STR, Wself, n0, ln, hi, FF);
    v8f hn = wave_gemm16(lx, STR, Wnb,   n0, ln, hi, FF);
    store_frag(hs, lhs, STR, h_self, row0, bself, n0, ln, hi);
    store_frag(hn, lhn, STR, h_nb,   row0, bnb,   n0, ln, hi);
    __syncthreads();

    // Stage 2: s = h_self @ A1_i, t = h_nb @ A1_j (chained through LDS)
    v8f sv = wave_gemm16(lhs, STR, A1,            n0, ln, hi, FF);
    v8f tv = wave_gemm16(lhn, STR, A1 + FF * FF,  n0, ln, hi, FF);
    store_frag(sv, nullptr, 0, s_out, row0, nullptr, n0, ln, hi);
    store_frag(tv, nullptr, 0, t_out, row0, nullptr, n0, ln, hi);
}

__global__ __launch_bounds__(256) void gat_attn_kernel(
    const float* __restrict__ s, const float* __restrict__ t,
    const float* __restrict__ h_self, const float* __restrict__ h_nb,
    const float* __restrict__ adj,
    const float* __restrict__ A1b, const float* __restrict__ A2w,
    const float* __restrict__ A2b,
    float* __restrict__ out)
{
    const int SSTR = 132;   // s tile stride
    const int TSTR = 133;   // t tile stride: banks (5*j + k) % 64, distinct for j=0..31
    const int CSTR = 516;   // score row stride: banks (4*i + k) % 64 for WMMA A-frags
    __shared__ float ls[16 * SSTR];
    __shared__ float lt[32 * TSTR];
    __shared__ float lsc[16 * CSTR];
    __shared__ float la1b[FF];
    __shared__ float la2[FF];
    __shared__ float red[256];
    __shared__ float rowstat[16];

    const int b  = blockIdx.x >> 5;     // 32 row-tiles per batch
    const int i0 = (blockIdx.x & 31) * 16;
    const float* sb   = s      + (size_t)b * NN * FF;
    const float* tb   = t      + (size_t)b * NN * FF;
    const float* hnb  = h_nb   + (size_t)b * NN * FF;
    const float* hsb  = h_self + (size_t)b * NN * FF;
    const float* adjb = adj    + (size_t)b * NN * NN;
    const int tid = threadIdx.x;

    for (int e = tid; e < 16 * FF; e += 256) {
        int r = e >> 7, c = e & 127;
        ls[r * SSTR + c] = sb[(i0 + r) * FF + c];
    }
    if (tid < FF) { la1b[tid] = A1b[tid]; la2[tid] = A2w[tid]; }
    const float a2bias = A2b[0];
    __syncthreads();

    // ---- Phase A: masked scores into LDS (relu-dot, not expressible as matmul) ----
    const int j5 = tid & 31;   // j within 32-wide tile (wave-uniform i -> LDS bcast)
    const int i8 = tid >> 5;   // 0..7; each thread also handles i8+8
    for (int j0 = 0; j0 < NN; j0 += 32) {
        for (int e = tid; e < 32 * FF; e += 256) {
            int r = e >> 7, c = e & 127;
            lt[r * TSTR + c] = tb[(j0 + r) * FF + c];
        }
        __syncthreads();
        #pragma unroll
        for (int p = 0; p < 2; ++p) {
            int i = i8 + 8 * p;
            float acc = 0.f;
            for (int k = 0; k < FF; ++k) {
                float pre = ls[i * SSTR + k] + lt[j5 * TSTR + k] + la1b[k];
                acc = fmaf(fmaxf(pre, 0.f), la2[k], acc);
            }
            acc += a2bias;
            float av = adjb[(i0 + i) * NN + (j0 + j5)];
            lsc[i * CSTR + j0 + j5] = (av == 0.f) ? -1.0e9f : acc;
        }
        __syncthreads();
    }

    // ---- Phase B: row softmax in LDS (16 threads per row) ----
    {
        const int row = tid >> 4, l = tid & 15;
        float m = -3.4e38f;
        for (int j = l; j < NN; j += 16) m = fmaxf(m, lsc[row * CSTR + j]);
        red[tid] = m;
        __syncthreads();
        if (l == 0) {
            float mm = red[row * 16];
            for (int q = 1; q < 16; ++q) mm = fmaxf(mm, red[row * 16 + q]);
            rowstat[row] = mm;
        }
        __syncthreads();
        const float rm = rowstat[row];
        float sum = 0.f;
        for (int j = l; j < NN; j += 16) {
            float e = __expf(lsc[row * CSTR + j] - rm);
            lsc[row * CSTR + j] = e;
            sum += e;
        }
        red[tid] = sum;
        __syncthreads();
        if (l == 0) {
            float ss = 0.f;
            for (int q = 0; q < 16; ++q) ss += red[row * 16 + q];
            rowstat[row] = 1.0f / ss;
        }
        __syncthreads();
        const float inv = rowstat[row];
        for (int j = l; j < NN; j += 16) lsc[row * CSTR + j] *= inv;
    }
    __syncthreads();

    // ---- Phase C: agg = attn @ h_nb via WMMA (A from LDS, K = 512), fused epilogue ----
    const int lane = tid & 31;
    const int ln = lane & 15, hi = lane >> 4;
    const int n0 = (tid >> 5) * 16;
    v8f acc = {0.f, 0.f, 0.f, 0.f, 0.f, 0.f, 0.f, 0.f};
    for (int k0 = 0; k0 < NN; k0 += 4) {
        int ka = k0 + 2 * hi;
        v2f a, bf;
        a.x  = lsc[ln * CSTR + ka];
        a.y  = lsc[ln * CSTR + ka + 1];
        bf.x = hnb[ka * FF + n0 + ln];
        bf.y = hnb[(ka + 1) * FF + n0 + ln];
        acc = __builtin_amdgcn_wmma_f32_16x16x4_f32(false, a, false, bf,
                                                    (short)0, acc, false, false);
    }
    #pragma unroll
    for (int v = 0; v < 8; ++v) {
        int r = v + 8 * hi;
        int c = n0 + ln;
        float o = acc[v] + hsb[(i0 + r) * FF + c];
        out[((size_t)b * NN + i0 + r) * FF + c] = fmaxf(o, 0.f);
    }
}

extern "C" void kernel_launch(void* const* d_in, const int* in_sizes, int n_in,
                              void* d_out, int out_size, void* d_ws, size_t ws_size,
                              hipStream_t stream) {
    (void)in_sizes; (void)n_in; (void)out_size; (void)ws_size;
    const float* X     = (const float*)d_in[0];   // (B,N,F_IN)
    const float* adj   = (const float*)d_in[1];   // (B,N,N)
    const float* Wself = (const float*)d_in[2];   // (F_IN,F_OUT)
    const float* bself = (const float*)d_in[3];   // (F_OUT)
    const float* Wnb   = (const float*)d_in[4];
    const float* bnb   = (const float*)d_in[5];
    const float* A1w   = (const float*)d_in[6];   // (2F,F)
    const float* A1b   = (const float*)d_in[7];   // (F)
    const float* A2w   = (const float*)d_in[8];   // (F,1)
    const float* A2b   = (const float*)d_in[9];   // (1)

    float* ws     = (float*)d_ws;
    const size_t HSZ = (size_t)NB * NN * FF;      // 262144 floats per buffer
    float* h_self = ws;
    float* h_nb   = ws + HSZ;
    float* s_buf  = ws + 2 * HSZ;
    float* t_buf  = ws + 3 * HSZ;

    gat_proj_kernel<<<(NB * NN) / 16, 256, 0, stream>>>(
        X, Wself, bself, Wnb, bnb, A1w, h_self, h_nb, s_buf, t_buf);
    gat_attn_kernel<<<NB * (NN / 16), 256, 0, stream>>>(
        s_buf, t_buf, h_self, h_nb, adj, A1b, A2w, A2b, (float*)d_out);
}